// GRAPH_VAE_V3_62663572849389
// MI455X (gfx1250) — compile-verified
//
#include <hip/hip_runtime.h>
#include <hip/hip_bf16.h>

typedef __attribute__((ext_vector_type(2))) float v2f;
typedef __attribute__((ext_vector_type(8))) float v8f;

#define NN   50000
#define EEDG 600000
#define DLAT 64

// ---------------- degree / normalization ----------------
__global__ void zero_i32_kernel(int* __restrict__ p, int n) {
    int i = blockIdx.x * blockDim.x + threadIdx.x;
    if (i < n) p[i] = 0;
}

__global__ void deg_count_kernel(const int* __restrict__ dst, int* __restrict__ deg) {
    int e = blockIdx.x * blockDim.x + threadIdx.x;
    if (e < EEDG) atomicAdd(&deg[dst[e]], 1);
}

__global__ void deg_finalize_kernel(const int* __restrict__ deg, float* __restrict__ dis) {
    int i = blockIdx.x * blockDim.x + threadIdx.x;
    if (i < NN) dis[i] = rsqrtf((float)deg[i] + 1.0f);
}

// ---------------- FP32 WMMA GEMM: h = X@W ; out = h*dis^2 + bias ----------------
// One wave32 computes one 16x16 tile of the output, K stepped by 4 through
// V_WMMA_F32_16X16X4_F32 (native fp32 matrix pipe; no precision loss vs reference).
__global__ void gemm_wmma_init_kernel(const float* __restrict__ X,
                                      const float* __restrict__ W,
                                      const float* __restrict__ bias,
                                      const float* __restrict__ dis,
                                      float* __restrict__ h,
                                      float* __restrict__ out,
                                      int K, int C) {
    int wave = blockIdx.x * (blockDim.x >> 5) + (threadIdx.x >> 5);
    int lane = threadIdx.x & 31;
    int tilesN = C >> 4;
    int tiles  = (NN >> 4) * tilesN;
    if (wave >= tiles) return;          // whole-wave uniform exit (EXEC all-1 for WMMA)

    int tm = wave / tilesN;
    int tn = wave - tm * tilesN;
    int row0 = tm << 4;
    int col0 = tn << 4;
    int half = lane >> 4;               // 0: lanes 0-15, 1: lanes 16-31
    int l    = lane & 15;

    // A (16x4 fp32): lane l+16*half holds X[row0+l][k+2*half .. k+2*half+1]
    const float* xr = X + (size_t)(row0 + l) * K + 2 * half;
    // B (4x16 fp32): lane l+16*half holds W[k+2*half .. k+2*half+1][col0+l]
    const float* wc = W + (size_t)(2 * half) * C + col0 + l;

    v8f acc = {0.f, 0.f, 0.f, 0.f, 0.f, 0.f, 0.f, 0.f};
    for (int k = 0; k < K; k += 4) {
        v2f a; a.x = xr[k];               a.y = xr[k + 1];
        v2f b; b.x = wc[(size_t)k * C];   b.y = wc[(size_t)(k + 1) * C];
        acc = __builtin_amdgcn_wmma_f32_16x16x4_f32(
            /*neg_a=*/false, a, /*neg_b=*/false, b,
            /*c_mod=*/(short)0, acc, /*reuse_a=*/false, /*reuse_b=*/false);
    }

    // C/D layout: VGPR r -> row (row0 + r + 8*half), col (col0 + l)
    int col = col0 + l;
    float bv = bias[col];
#pragma unroll
    for (int r = 0; r < 8; ++r) {
        int row = row0 + r + 8 * half;
        float v = acc[r];
        size_t o = (size_t)row * C + col;
        h[o] = v;                         // raw x@W (needed for edge gather)
        float d = dis[row];
        out[o] = v * d * d + bv;          // self-loop term + bias, pre-scatter init
    }
}

// ---------------- edge scatter: out[dst] += h[src] * dis[src]*dis[dst] ----------------
__global__ void edge_scatter_kernel(const int* __restrict__ src,
                                    const int* __restrict__ dst,
                                    const float* __restrict__ dis,
                                    const float* __restrict__ h,
                                    float* __restrict__ out,
                                    int cshift) {
    long long idx = (long long)blockIdx.x * blockDim.x + threadIdx.x;
    int e = (int)(idx >> cshift);
    if (e >= EEDG) return;
    int C = 1 << cshift;
    int c = (int)idx & (C - 1);
    int s = src[e];
    int d = dst[e];
    float nrm = dis[s] * dis[d];
    unsafeAtomicAdd(&out[(size_t)d * C + c], h[(size_t)s * C + c] * nrm);
}

__global__ void relu_kernel(float* __restrict__ p, int n) {
    int i = blockIdx.x * blockDim.x + threadIdx.x;
    if (i < n) p[i] = fmaxf(p[i], 0.0f);
}

// ---------------- reparameterize: clamp logstd, z = mu + eps*exp(ls) ----------------
__global__ void reparam_kernel(const float* __restrict__ mu,
                               const float* __restrict__ ls_act,
                               const float* __restrict__ eps,
                               float* __restrict__ ls_out,
                               float* __restrict__ z) {
    int i = blockIdx.x * blockDim.x + threadIdx.x;
    if (i < NN * DLAT) {
        float l = fminf(ls_act[i], 10.0f);
        ls_out[i] = l;
        z[i] = mu[i] + eps[i] * __expf(l);
    }
}

// ---------------- decoder: adj[e] = sigmoid(dot(z[src], z[dst])) ----------------
__global__ void decode_kernel(const int* __restrict__ src,
                              const int* __restrict__ dst,
                              const float* __restrict__ z,
                              float* __restrict__ adj) {
    int e = blockIdx.x * blockDim.x + threadIdx.x;
    if (e >= EEDG) return;
    const float4* a = (const float4*)(z + (size_t)src[e] * DLAT);
    const float4* b = (const float4*)(z + (size_t)dst[e] * DLAT);
    float acc = 0.0f;
#pragma unroll
    for (int i = 0; i < DLAT / 4; ++i) {
        float4 x = a[i];
        float4 y = b[i];
        acc += x.x * y.x + x.y * y.y + x.z * y.z + x.w * y.w;
    }
    adj[e] = 1.0f / (1.0f + __expf(-acc));
}

extern "C" void kernel_launch(void* const* d_in, const int* in_sizes, int n_in,
                              void* d_out, int out_size, void* d_ws, size_t ws_size,
                              hipStream_t stream) {
    (void)in_sizes; (void)n_in; (void)out_size; (void)ws_size;

    const float* x   = (const float*)d_in[0];
    const int*   ei  = (const int*)d_in[1];
    const int*   src = ei;
    const int*   dst = ei + EEDG;
    const float* eps = (const float*)d_in[2];
    const float* w[2][3] = {
        {(const float*)d_in[3], (const float*)d_in[5], (const float*)d_in[7]},
        {(const float*)d_in[9], (const float*)d_in[11], (const float*)d_in[13]}};
    const float* bb[2][3] = {
        {(const float*)d_in[4], (const float*)d_in[6], (const float*)d_in[8]},
        {(const float*)d_in[10], (const float*)d_in[12], (const float*)d_in[14]}};

    float* out    = (float*)d_out;
    float* adj    = out;                          // [E]
    float* mu_out = out + EEDG;                   // [N, 64]
    float* ls_out = mu_out + (size_t)NN * DLAT;   // [N, 64]

    // workspace layout (floats): dis[N] | h[N*128] | bufA[N*128] | bufB[N*128]
    float* ws   = (float*)d_ws;
    float* dis  = ws;
    float* hbuf = dis + NN;
    float* bufA = hbuf + (size_t)NN * 128;
    float* bufB = bufA + (size_t)NN * 128;
    float* zbuf = hbuf;            // h is dead by reparam time -> reuse
    int*   deg  = (int*)bufA;      // deg only live before bufA is first written

    const int T = 256;

    // --- degree + rsqrt normalization ---
    zero_i32_kernel<<<(NN + T - 1) / T, T, 0, stream>>>(deg, NN);
    deg_count_kernel<<<(EEDG + T - 1) / T, T, 0, stream>>>(dst, deg);
    deg_finalize_kernel<<<(NN + T - 1) / T, T, 0, stream>>>(deg, dis);

    const int dims_in[3]  = {128, 64, 128};
    const int dims_out[3] = {64, 128, 64};

    for (int br = 0; br < 2; ++br) {
        const float* cur = x;
        for (int li = 0; li < 3; ++li) {
            int K = dims_in[li];
            int C = dims_out[li];
            float* o;
            if (li == 0)      o = bufA;
            else if (li == 1) o = bufB;
            else              o = (br == 0) ? mu_out : bufA;

            // GEMM (fp32 WMMA) + self-loop/bias init
            int tiles  = (NN / 16) * (C / 16);
            int blocks = (tiles + 7) / 8;        // 8 waves of 32 per block
            gemm_wmma_init_kernel<<<blocks, T, 0, stream>>>(cur, w[br][li], bb[br][li],
                                                            dis, hbuf, o, K, C);
            // edge scatter (atomic fp32 adds; working set fits in 192MB L2)
            int cshift = (C == 128) ? 7 : 6;
            long long tot = (long long)EEDG << cshift;
            edge_scatter_kernel<<<(unsigned)((tot + T - 1) / T), T, 0, stream>>>(
                src, dst, dis, hbuf, o, cshift);
            // ReLU
            int n = NN * C;
            relu_kernel<<<(n + T - 1) / T, T, 0, stream>>>(o, n);
            cur = o;
        }
    }

    // mu in mu_out, logstd activation in bufA
    reparam_kernel<<<(NN * DLAT + T - 1) / T, T, 0, stream>>>(mu_out, bufA, eps,
                                                              ls_out, zbuf);
    decode_kernel<<<(EEDG + T - 1) / T, T, 0, stream>>>(src, dst, zbuf, adj);
}